// HyIE_88347477279160
// MI455X (gfx1250) — compile-verified
//
#include <hip/hip_runtime.h>
#include <math.h>

// ---------------- problem constants (from reference) ----------------
#define NUM_ENT   50000
#define NUM_RELN  500
#define KTOP      50
#define EMB       400          // D1*D2
#define FEAT      1000
#define N_NODES   50500        // NUM_ENT + NUM_REL
#define N_EDGES   400000
#define BATCH     4096
#define RELROWS   499          // NUM_REL - 1
#define RA_N      100          // 2*K
#define RA_E      2000

// GEMM blocking
#define BM     128             // rows per block (8 waves x 16)
#define NT     5               // 16-col C tiles per wave
#define BROW   (NT * 16)       // 80 columns per block
#define KC     32              // K chunk staged by TDM (dwords)
#define LDSROW (KC + 1)        // 33-dword A row stride via TDM pad (1 dword / 32)
#define NCHUNK 31              // 31*32 = 992
#define KTAIL  992             // remaining 8 k's done with direct loads

typedef __attribute__((ext_vector_type(2))) float v2f;
typedef __attribute__((ext_vector_type(8))) float v8f;
typedef unsigned int u32;
typedef unsigned long long u64;
typedef __attribute__((ext_vector_type(4))) u32 v4u;
typedef __attribute__((ext_vector_type(8))) u32 v8u;

#ifndef __has_builtin
#define __has_builtin(x) 0
#endif
#if defined(__gfx1250__) && __has_builtin(__builtin_amdgcn_wmma_f32_16x16x4_f32)
#define GFX1250_PATH 1
#else
#define GFX1250_PATH 0
#endif

#if GFX1250_PATH
// ---- Tensor Data Mover: 2D tile global -> LDS (D# per CDNA5 ISA,
// 2-group form, VADDR2/3 = NULL). padword packs pad_enable/interval/amount bits.
__device__ __forceinline__ void tdm_load_2d(u32 lds_off, u64 gaddr,
                                            u32 dim0, u32 dim1,
                                            u32 tile0, u32 tile1,
                                            u32 stride, u32 padword) {
  v4u g0;
  g0[0] = 1u;                                              // count=1 (valid), user mode
  g0[1] = lds_off;                                         // LDS byte address
  g0[2] = (u32)gaddr;                                      // global_addr[31:0]
  g0[3] = (u32)((gaddr >> 32) & 0x01ffffffu) | (2u << 30); // global_addr[56:32], type=2
  v8u g1;
  g1[0] = (2u << 16) | padword;                            // data_size=4B | pad bits
  g1[1] = (dim0 & 0xffffu) << 16;                          // tensor_dim0[15:0]
  g1[2] = (dim0 >> 16) | ((dim1 & 0xffffu) << 16);         // dim0[31:16] | dim1[15:0]
  g1[3] = (dim1 >> 16) | (tile0 << 16);                    // dim1[31:16] | tile_dim0
  g1[4] = tile1;                                           // tile_dim1, tile_dim2=0
  g1[5] = stride;                                          // tensor_dim0_stride[31:0]
  g1[6] = 0u;
  g1[7] = 0u;
  asm volatile("tensor_load_to_lds %0, %1" :: "s"(g0), "s"(g1) : "memory");
}
#define PADW_A ((1u << 20) | (4u << 22) | (0u << 25))      // pad 1 dword every 32
#define PADW_B 0u
#endif

// ---------------- small utility kernels ----------------
__global__ void k_zero2(float* a, int na, float* b, int nb) {
  int g = blockIdx.x * blockDim.x + threadIdx.x;
  if (g < na) a[g] = 0.f;
  if (g < nb) b[g] = 0.f;
}

__global__ void k_fin_dinv(float* d, int n) {
  int g = blockIdx.x * blockDim.x + threadIdx.x;
  if (g < n) d[g] = rsqrtf(d[g] + 1.0f);
}

__global__ void k_relu(float* x, int n) {
  int g = blockIdx.x * blockDim.x + threadIdx.x;
  if (g < n) x[g] = fmaxf(x[g], 0.f);
}

__global__ void k_copy4(const float4* __restrict__ src, float4* __restrict__ dst, int n4) {
  int g = blockIdx.x * blockDim.x + threadIdx.x;
  if (g < n4) dst[g] = src[g];
}

// out[m][n] = in[m][n] * dinv[m]^2 + bias[n]
__global__ void k_init_scaled(const float* __restrict__ in, const float* __restrict__ dinv,
                              const float* __restrict__ bias, float* __restrict__ out,
                              int ntot, int ncols) {
  int g = blockIdx.x * blockDim.x + threadIdx.x;
  if (g >= ntot) return;
  int m = g / ncols, n = g % ncols;
  float s = dinv[m];
  out[g] = in[g] * s * s + bias[n];
}

__global__ void k_deg(const int* __restrict__ ei, int nedges, float* dinv) {
  int e = blockIdx.x * blockDim.x + threadIdx.x;
  if (e < nedges) atomicAdd(&dinv[ei[nedges + e]], 1.0f);
}

__global__ void k_scatter4(const int* __restrict__ ei, int nedges,
                           const float* __restrict__ xw, const float* __restrict__ dinv,
                           float* __restrict__ out, int ncols4, int ncols) {
  int g = blockIdx.x * blockDim.x + threadIdx.x;
  int total = nedges * ncols4;
  if (g >= total) return;
  int e = g / ncols4, ch = g % ncols4;
  int s = ei[e], d = ei[nedges + e];
  float w = dinv[s] * dinv[d];
  float4 v = ((const float4*)(xw + (size_t)s * ncols))[ch];
  float* o = out + (size_t)d * ncols + (size_t)ch * 4;
  atomicAdd(o + 0, v.x * w);
  atomicAdd(o + 1, v.y * w);
  atomicAdd(o + 2, v.z * w);
  atomicAdd(o + 3, v.w * w);
}

// ---------------- stage 1: relation attention + top-k ----------------
__global__ __launch_bounds__(256)
void k_logits(const float* __restrict__ nf, const float* __restrict__ Wp,
              const float* __restrict__ bp, float* __restrict__ prob) {
  __shared__ float red[256];
  int row = blockIdx.x;
  float s = 0.f;
  for (int c = threadIdx.x; c < FEAT; c += 256)
    s += nf[(size_t)row * FEAT + c] * Wp[c];
  red[threadIdx.x] = s; __syncthreads();
  for (int off = 128; off > 0; off >>= 1) {
    if (threadIdx.x < off) red[threadIdx.x] += red[threadIdx.x + off];
    __syncthreads();
  }
  if (threadIdx.x == 0) prob[row] = red[0] + bp[0];
}

__global__ __launch_bounds__(512)
void k_softmax_topk(float* __restrict__ prob, int* __restrict__ idx) {
  __shared__ float l[RELROWS];
  __shared__ int flag[RELROWS];
  __shared__ float mx, sm;
  int t = threadIdx.x;
  for (int i = t; i < RELROWS; i += 512) l[i] = prob[i];
  __syncthreads();
  if (t == 0) { float m = -1e30f; for (int i = 0; i < RELROWS; ++i) m = fmaxf(m, l[i]); mx = m; }
  __syncthreads();
  for (int i = t; i < RELROWS; i += 512) l[i] = expf(l[i] - mx);
  __syncthreads();
  if (t == 0) { float s = 0.f; for (int i = 0; i < RELROWS; ++i) s += l[i]; sm = s; }
  __syncthreads();
  for (int i = t; i < RELROWS; i += 512) { float p = l[i] / sm; l[i] = p; prob[i] = p; }
  __syncthreads();
  for (int i = t; i < RELROWS; i += 512) {      // jax top_k tie-break: lower index wins
    float pi = l[i]; int r = 0;
    for (int j = 0; j < RELROWS; ++j) {
      float pj = l[j];
      r += (pj > pi) || (pj == pi && j < i);
    }
    flag[i] = (r < KTOP) ? 1 : 0;
  }
  __syncthreads();
  if (t == 0) {
    int pos = 0;
    for (int i = 0; i < RELROWS && pos < KTOP; ++i)
      if (flag[i]) idx[pos++] = i;              // ascending -> already sorted
  }
}

__global__ void k_build_ra(const float* __restrict__ nf, const float* __restrict__ anchor,
                           const float* __restrict__ prob, const int* __restrict__ idx,
                           float* __restrict__ ra) {
  int g = blockIdx.x * blockDim.x + threadIdx.x;
  if (g >= RA_N * FEAT) return;
  int r = g / FEAT, c = g % FEAT;
  float v;
  if (r < KTOP) v = anchor[(size_t)r * FEAT + c];
  else { int i = idx[r - KTOP]; v = nf[(size_t)i * FEAT + c] * prob[i]; }
  ra[g] = v;
}

// tiny GEMM: xw1(100x1000) = ra(100x1000) @ Wa(1000x1000)  (1e8 FMA, negligible)
__global__ __launch_bounds__(256)
void k_gemm_small(const float* __restrict__ ra, const float* __restrict__ Wa,
                  float* __restrict__ xw1) {
  int g = blockIdx.x * blockDim.x + threadIdx.x;
  if (g >= RA_N * FEAT) return;
  int m = g / FEAT, n = g % FEAT;
  const float* ar = ra + (size_t)m * FEAT;
  float s = 0.f;
  for (int k = 0; k < FEAT; ++k) s += ar[k] * Wa[(size_t)k * FEAT + n];
  xw1[g] = s;
}

__global__ void k_temp_scatter(const int* __restrict__ idx, const float* __restrict__ x1,
                               float* __restrict__ temp) {
  int g = blockIdx.x * blockDim.x + threadIdx.x;
  if (g >= KTOP * FEAT) return;
  int j = g / FEAT, c = g % FEAT;
  temp[(size_t)idx[j] * FEAT + c] = x1[(size_t)(KTOP + j) * FEAT + c];
}

// ---------------- main GEMM: C(M x 400) = A(M x 1000) @ B(1000 x 400), fp32 WMMA ----------------
// Block: 8 waves = 128 rows x 80 cols. TDM double-buffers BOTH operands into LDS:
//   wave0 owns the A descriptor (128xKC, padded rows), wave1 owns the B descriptor
//   (KCx80, natural 80-dword rows -> conflict-free b-fragment reads).
// Inner loop is pure LDS (ds_load_2addr) -> v_wmma_f32_16x16x4_f32, 40 wmma/chunk/wave.
// Edge rows handled by tensor_dim clamp (TDM returns zeros past tensor extent).
__global__ __launch_bounds__(256)
void k_gemm_wmma(const float* __restrict__ A, const float* __restrict__ B,
                 float* __restrict__ C, int M, int N) {
  const int lane  = threadIdx.x & 31;
  const int wave  = threadIdx.x >> 5;
  const int row0  = blockIdx.y * BM;
  const int col0  = blockIdx.x * BROW;
  const int r     = (wave << 4) + (lane & 15);   // row within 128-row tile
  const int khalf = (lane >> 4) << 1;            // 0 or 2
  const int nloc  = lane & 15;

#if GFX1250_PATH
  __shared__ float atile[2][BM * LDSROW];        // 33792 B
  __shared__ float btile[2][KC * BROW];          // 20480 B

  const u32 ldsA0 = (u32)(uintptr_t)&atile[0][0];
  const u32 ldsA1 = (u32)(uintptr_t)&atile[1][0];
  const u32 ldsB0 = (u32)(uintptr_t)&btile[0][0];
  const u32 ldsB1 = (u32)(uintptr_t)&btile[1][0];
  const u32 dimA1 = (u32)((M - row0 < BM) ? (M - row0) : BM);
  const u64 abase = (u64)(uintptr_t)A + (u64)row0 * FEAT * 4u;
  const u64 bbase = (u64)(uintptr_t)B + (u64)col0 * 4u;

  if (wave == 0)
    tdm_load_2d(ldsA0, abase, (u32)FEAT, dimA1, KC, BM, (u32)FEAT, PADW_A);
  else if (wave == 1)
    tdm_load_2d(ldsB0, bbase, (u32)(N - col0), (u32)FEAT, BROW, KC, (u32)N, PADW_B);

  v8f acc[NT];
  #pragma unroll
  for (int t = 0; t < NT; ++t) acc[t] = (v8f){};

  for (int c = 0; c < NCHUNK; ++c) {
    if (wave == 0) {
      if (c + 1 < NCHUNK) {
        tdm_load_2d((c & 1) ? ldsA0 : ldsA1, abase + (u64)(c + 1) * KC * 4u,
                    (u32)(FEAT - (c + 1) * KC), dimA1, KC, BM, (u32)FEAT, PADW_A);
        __builtin_amdgcn_s_wait_tensorcnt(1);    // A chunk c complete, c+1 in flight
      } else {
        __builtin_amdgcn_s_wait_tensorcnt(0);
      }
    } else if (wave == 1) {
      if (c + 1 < NCHUNK) {
        tdm_load_2d((c & 1) ? ldsB0 : ldsB1,
                    bbase + (u64)(c + 1) * KC * N * 4u,
                    (u32)(N - col0), (u32)(FEAT - (c + 1) * KC), BROW, KC, (u32)N, PADW_B);
        __builtin_amdgcn_s_wait_tensorcnt(1);    // B chunk c complete, c+1 in flight
      } else {
        __builtin_amdgcn_s_wait_tensorcnt(0);
      }
    }
    __syncthreads();                             // chunk c tiles visible to all waves
    const float* as = &atile[c & 1][r * LDSROW];
    const float* bs = &btile[c & 1][0];
    #pragma unroll
    for (int kk = 0; kk < KC; kk += 4) {
      v2f a;
      a.x = as[kk + khalf];
      a.y = as[kk + khalf + 1];
      #pragma unroll
      for (int t = 0; t < NT; ++t) {
        const float* bb = bs + (kk + khalf) * BROW + t * 16 + nloc;
        v2f b; b.x = bb[0]; b.y = bb[BROW];
        acc[t] = __builtin_amdgcn_wmma_f32_16x16x4_f32(false, a, false, b,
                                                       (short)0, acc[t], false, false);
      }
    }
    __syncthreads();                             // reads done before buffer reuse
  }

  // K tail (992..1000): direct global loads, row pointer clamped (edge rows
  // compute garbage but are never stored).
  {
    const int m = row0 + r;
    const float* arow = A + (size_t)(m < M ? m : 0) * FEAT;
    #pragma unroll
    for (int kk = KTAIL; kk < FEAT; kk += 4) {
      v2f a; a.x = arow[kk + khalf]; a.y = arow[kk + khalf + 1];
      #pragma unroll
      for (int t = 0; t < NT; ++t) {
        const float* bp = B + (size_t)(kk + khalf) * N + (col0 + t * 16 + nloc);
        v2f b; b.x = bp[0]; b.y = bp[N];
        acc[t] = __builtin_amdgcn_wmma_f32_16x16x4_f32(false, a, false, b,
                                                       (short)0, acc[t], false, false);
      }
    }
  }

  // store: vgpr j -> M = mb + j, N = col (C layout: M = j + 8*(lane>=16))
  const int mb = row0 + (wave << 4) + ((lane >> 4) << 3);
  #pragma unroll
  for (int t = 0; t < NT; ++t) {
    int col = col0 + t * 16 + nloc;
    #pragma unroll
    for (int j = 0; j < 8; ++j) {
      int mm = mb + j;
      if (mm < M) C[(size_t)mm * N + col] = acc[t][j];
    }
  }
#else
  // portable fallback (host pass / non-gfx1250): plain dot products
  const int m = row0 + r;
  if (m < M) {
    for (int t = 0; t < NT; ++t) {
      int col = col0 + t * 16 + nloc;
      float s0 = 0.f;
      for (int k = 0; k < FEAT; ++k) s0 += A[(size_t)m * FEAT + k] * B[(size_t)k * N + col];
      if ((lane >> 4) == 0) C[(size_t)m * N + col] = s0;
    }
  }
#endif
}

// ---------------- ConvE-style scoring ----------------
__global__ __launch_bounds__(256)
void k_score(const float* __restrict__ ent, const float* __restrict__ rel,
             const float* __restrict__ cw, const float* __restrict__ cb,
             const float* __restrict__ fcw, const float* __restrict__ fcb,
             const int* __restrict__ ridx, const int* __restrict__ e1i,
             const int* __restrict__ e2i, float* __restrict__ score) {
  __shared__ float red[256];
  int b = blockIdx.x;
  const float* rr = rel + (size_t)ridx[b] * EMB;
  const float* a1 = ent + (size_t)e1i[b] * EMB;
  const float* a2 = ent + (size_t)e2i[b] * EMB;
  float s = 0.f;
  for (int pos = threadIdx.x; pos < EMB; pos += 256) {
    float rv = rr[pos], v1 = a1[pos], v2 = a2[pos];
    #pragma unroll
    for (int p = 0; p < 4; ++p) {
      int o0 = 2 * p, o1 = 2 * p + 1;
      float h0 = cw[o0*3+0]*rv + cw[o0*3+1]*v1 + cw[o0*3+2]*v2 + cb[o0];
      float h1 = cw[o1*3+0]*rv + cw[o1*3+1]*v1 + cw[o1*3+2]*v2 + cb[o1];
      s += fmaxf(h0, h1) * fcw[p * EMB + pos];
    }
  }
  red[threadIdx.x] = s; __syncthreads();
  for (int off = 128; off > 0; off >>= 1) {
    if (threadIdx.x < off) red[threadIdx.x] += red[threadIdx.x + off];
    __syncthreads();
  }
  if (threadIdx.x == 0) score[b] = red[0] + fcb[0];
}

// ---------------- launch ----------------
extern "C" void kernel_launch(void* const* d_in, const int* in_sizes, int n_in,
                              void* d_out, int out_size, void* d_ws, size_t ws_size,
                              hipStream_t stream) {
  const float* num_feature = (const float*)d_in[0];
  const float* anchor      = (const float*)d_in[1];
  const float* Wp          = (const float*)d_in[2];
  const float* bp          = (const float*)d_in[3];
  const float* Wa          = (const float*)d_in[4];
  const float* ba          = (const float*)d_in[5];
  const float* W1          = (const float*)d_in[6];
  const float* b1          = (const float*)d_in[7];
  const float* ent_emb     = (const float*)d_in[8];
  const float* rel_emb     = (const float*)d_in[9];
  const float* conv_w      = (const float*)d_in[10];
  const float* conv_b      = (const float*)d_in[11];
  const float* fc_w        = (const float*)d_in[12];
  const float* fc_b        = (const float*)d_in[13];
  const int*   edge_index  = (const int*)d_in[14];
  const int*   rel_ei      = (const int*)d_in[15];
  const int*   r_idx       = (const int*)d_in[16];
  const int*   e1_idx      = (const int*)d_in[17];
  const int*   e2_idx      = (const int*)d_in[18];

  // outputs: score(4096) | x(50500*400) | temp(50500*1000)
  float* out   = (float*)d_out;
  float* score = out;
  float* x_out = out + BATCH;
  float* temp  = out + BATCH + (size_t)N_NODES * EMB;

  // workspace layout (floats)
  float* XW    = (float*)d_ws;               // N_NODES*EMB
  float* DINV  = XW + (size_t)N_NODES * EMB; // N_NODES
  float* PROB  = DINV + N_NODES;             // 512
  float* RAB   = PROB + 512;                 // RA_N*FEAT
  float* XW1   = RAB + RA_N * FEAT;          // RA_N*FEAT
  float* X1    = XW1 + RA_N * FEAT;          // RA_N*FEAT
  float* DINV1 = X1 + RA_N * FEAT;           // 128
  int*   IDX   = (int*)(DINV1 + 128);        // 64

  const int T = 256;

  // 0) zero degree accumulators
  k_zero2<<<(N_NODES + T - 1) / T, T, 0, stream>>>(DINV, N_NODES, DINV1, RA_N);

  // 1) relation attention: logits -> softmax -> top-50 (sorted)
  k_logits<<<RELROWS, T, 0, stream>>>(num_feature, Wp, bp, PROB);
  k_softmax_topk<<<1, 512, 0, stream>>>(PROB, IDX);

  // 2) ra = concat(anchor, rel_feat[idx]*prob[idx]); GCN over 100-node graph
  k_build_ra<<<(RA_N * FEAT + T - 1) / T, T, 0, stream>>>(num_feature, anchor, PROB, IDX, RAB);
  k_gemm_small<<<(RA_N * FEAT + T - 1) / T, T, 0, stream>>>(RAB, Wa, XW1);
  k_deg<<<(RA_E + T - 1) / T, T, 0, stream>>>(rel_ei, RA_E, DINV1);
  k_fin_dinv<<<1, 128, 0, stream>>>(DINV1, RA_N);
  k_init_scaled<<<(RA_N * FEAT + T - 1) / T, T, 0, stream>>>(XW1, DINV1, ba, X1, RA_N * FEAT, FEAT);
  k_scatter4<<<(RA_E * (FEAT / 4) + T - 1) / T, T, 0, stream>>>(rel_ei, RA_E, XW1, DINV1, X1, FEAT / 4, FEAT);
  k_relu<<<(RA_N * FEAT + T - 1) / T, T, 0, stream>>>(X1, RA_N * FEAT);

  // 3) temp = num_feature with rows[idx] <- x1[K:]
  {
    int n4 = (int)((size_t)N_NODES * FEAT / 4);
    k_copy4<<<(n4 + T - 1) / T, T, 0, stream>>>((const float4*)num_feature, (float4*)temp, n4);
    k_temp_scatter<<<(KTOP * FEAT + T - 1) / T, T, 0, stream>>>(IDX, X1, temp);
  }

  // 4) main GCN: xw = num_feature @ W1 via TDM-staged fp32 WMMA
  {
    dim3 grid(EMB / BROW, (N_NODES + BM - 1) / BM);        // (5, 395)
    k_gemm_wmma<<<grid, 256, 0, stream>>>(num_feature, W1, XW, N_NODES, EMB);
  }
  k_deg<<<(N_EDGES + T - 1) / T, T, 0, stream>>>(edge_index, N_EDGES, DINV);
  k_fin_dinv<<<(N_NODES + T - 1) / T, T, 0, stream>>>(DINV, N_NODES);
  {
    int ntot = N_NODES * EMB;  // 20.2M
    k_init_scaled<<<(ntot + T - 1) / T, T, 0, stream>>>(XW, DINV, b1, x_out, ntot, EMB);
    k_scatter4<<<(N_EDGES * (EMB / 4) + T - 1) / T, T, 0, stream>>>(edge_index, N_EDGES, XW, DINV, x_out, EMB / 4, EMB);
    k_relu<<<(ntot + T - 1) / T, T, 0, stream>>>(x_out, ntot);
  }

  // 5) ConvE scoring
  k_score<<<BATCH, 256, 0, stream>>>(ent_emb, rel_emb, conv_w, conv_b, fc_w, fc_b,
                                     r_idx, e1_idx, e2_idx, score);
}